// TopKPoolingNet_8117488189555
// MI455X (gfx1250) — compile-verified
//
#include <hip/hip_runtime.h>
#include <math.h>

// ---------------------------------------------------------------------------
// TopKPoolingNet on MI455X (gfx1250).  fp32 throughout; dense GEMM stages use
// V_WMMA_F32_16X16X4_F32 (exact fp32 matrix op), edge aggregation uses
// global_atomic_add_f32 into L2.  Memory-bound workload (~3GB edge traffic).
// ---------------------------------------------------------------------------

typedef __attribute__((ext_vector_type(2))) float v2f;
typedef __attribute__((ext_vector_type(8))) float v8f;

#define NBINS 2048

__device__ __forceinline__ float gelu_exact(float v) {
    return 0.5f * v * (1.0f + erff(v * 0.70710678118654752f));
}

__device__ __forceinline__ void atom_add_f32(float* p, float v) {
    unsafeAtomicAdd(p, v);   // -> global_atomic_add_f32
}

// --------------------------- small dense input -----------------------------
// h0 = gelu(x @ wd_root + wd_b)   [N,64] x [64,4] -> [N,4]
__global__ void dense_input_kernel(const float* __restrict__ x,
                                   const float* __restrict__ wd,
                                   const float* __restrict__ bd,
                                   float* __restrict__ h0, int n) {
    int i = blockIdx.x * blockDim.x + threadIdx.x;
    if (i >= n) return;
    float acc0 = bd[0], acc1 = bd[1], acc2 = bd[2], acc3 = bd[3];
    const float* xr = x + i * 64;
    #pragma unroll 8
    for (int f = 0; f < 64; ++f) {
        float xv = xr[f];
        const float* wr = wd + f * 4;
        acc0 += xv * wr[0]; acc1 += xv * wr[1];
        acc2 += xv * wr[2]; acc3 += xv * wr[3];
    }
    float* o = h0 + i * 4;
    o[0] = gelu_exact(acc0); o[1] = gelu_exact(acc1);
    o[2] = gelu_exact(acc2); o[3] = gelu_exact(acc3);
}

// --------------------------- edge aggregation ------------------------------
// agg[dst] += h[src], F floats per node, float4 granularity.
template <int F>
__global__ void edge_agg_kernel(const int* __restrict__ src,
                                const int* __restrict__ dst,
                                const float* __restrict__ h,
                                float* __restrict__ agg, int E) {
    const int C4 = F / 4;
    long long idx = (long long)blockIdx.x * blockDim.x + threadIdx.x;
    if (idx >= (long long)E * C4) return;
    int e = (int)(idx / C4);
    int c = (int)(idx % C4) * 4;
    int s = src[e], d = dst[e];
    float4 v = *(const float4*)(h + (long long)s * F + c);
    float* p = agg + (long long)d * F + c;
    atom_add_f32(p + 0, v.x); atom_add_f32(p + 1, v.y);
    atom_add_f32(p + 2, v.z); atom_add_f32(p + 3, v.w);
}

// filtered variant for the post-TopK graph (node_map remap + implicit mask)
__global__ void edge_agg_filtered_kernel(const int* __restrict__ src,
                                         const int* __restrict__ dst,
                                         const int* __restrict__ nmap,
                                         const float* __restrict__ hk,
                                         float* __restrict__ agg, int E) {
    long long idx = (long long)blockIdx.x * blockDim.x + threadIdx.x;
    if (idx >= (long long)E * 32) return;
    int e = (int)(idx >> 5);
    int c = (int)(idx & 31) * 4;
    int s = nmap[src[e]];
    int d = nmap[dst[e]];
    if (s < 0 || d < 0) return;
    float4 v = *(const float4*)(hk + (long long)s * 128 + c);
    float* p = agg + (long long)d * 128 + c;
    atom_add_f32(p + 0, v.x); atom_add_f32(p + 1, v.y);
    atom_add_f32(p + 2, v.z); atom_add_f32(p + 3, v.w);
}

// ------------------------ fused WMMA GraphConv GEMM ------------------------
// out = gelu(agg @ Wrel + x @ Wroot + bias), one 16x16 tile per wave,
// fp32 WMMA with K=4 steps.  A 16x4 layout: lane = row (lane&15),
// lane>>4 selects K-pair; VGPR0/1 = K+0/K+1.  B 4x16 mirrored on columns.
template <int FIN, int FOUT>
__global__ __launch_bounds__(128)
void graphconv_gemm_kernel(const float* __restrict__ agg,
                           const float* __restrict__ x,
                           const float* __restrict__ wrel,
                           const float* __restrict__ wroot,
                           const float* __restrict__ bias,
                           float* __restrict__ out, int rows) {
    const int nTiles = FOUT / 16;
    const int total = (rows / 16) * nTiles;
    int tile = blockIdx.x * 4 + (threadIdx.x >> 5);
    if (tile >= total) return;                 // wave-uniform, EXEC stays full
    const int mt = tile / nTiles, nt = tile % nTiles;
    const int lane = threadIdx.x & 31;
    const int r = lane & 15, kh = lane >> 4;
    const int m0 = mt * 16, n0 = nt * 16;

    v8f acc = {0.f, 0.f, 0.f, 0.f, 0.f, 0.f, 0.f, 0.f};

    const float* arow = agg + (long long)(m0 + r) * FIN + 2 * kh;
    #pragma unroll
    for (int k = 0; k < FIN; k += 4) {
        v2f a, b;
        a.x = arow[k + 0];
        a.y = arow[k + 1];
        const float* bp = wrel + (long long)(k + 2 * kh) * FOUT + n0 + r;
        b.x = bp[0];
        b.y = bp[FOUT];
        acc = __builtin_amdgcn_wmma_f32_16x16x4_f32(false, a, false, b,
                                                    (short)0, acc, false, false);
    }
    const float* xrow = x + (long long)(m0 + r) * FIN + 2 * kh;
    #pragma unroll
    for (int k = 0; k < FIN; k += 4) {
        v2f a, b;
        a.x = xrow[k + 0];
        a.y = xrow[k + 1];
        const float* bp = wroot + (long long)(k + 2 * kh) * FOUT + n0 + r;
        b.x = bp[0];
        b.y = bp[FOUT];
        acc = __builtin_amdgcn_wmma_f32_16x16x4_f32(false, a, false, b,
                                                    (short)0, acc, false, false);
    }

    // C/D layout: VGPR g -> row m0+g+8*kh, col n0+(lane&15)
    const int col = n0 + r;
    const float bv = bias[col];
    #pragma unroll
    for (int g = 0; g < 8; ++g) {
        int row = m0 + g + 8 * kh;
        out[(long long)row * FOUT + col] = gelu_exact(acc[g] + bv);
    }
}

// ------------------------------- TopK chain --------------------------------
__global__ void norm_kernel(const float* __restrict__ p, float* __restrict__ nrm) {
    __shared__ float s[128];
    int t = threadIdx.x;
    float v = p[t];
    s[t] = v * v;
    __syncthreads();
    for (int o = 64; o > 0; o >>= 1) {
        if (t < o) s[t] += s[t + o];
        __syncthreads();
    }
    if (t == 0) nrm[0] = sqrtf(s[0]);
}

__global__ void score_kernel(const float* __restrict__ h3,
                             const float* __restrict__ p,
                             const float* __restrict__ nrm,
                             float* __restrict__ scores, int n) {
    int i = blockIdx.x * blockDim.x + threadIdx.x;
    if (i >= n) return;
    const float* hr = h3 + (long long)i * 128;
    float acc = 0.f;
    #pragma unroll 8
    for (int f = 0; f < 128; ++f) acc += hr[f] * p[f];
    scores[i] = tanhf(acc / nrm[0]);
}

__global__ void hist_kernel(const float* __restrict__ scores,
                            int* __restrict__ hist,
                            int* __restrict__ bins, int n) {
    int i = blockIdx.x * blockDim.x + threadIdx.x;
    if (i >= n) return;
    int b = (int)((scores[i] + 1.0f) * (NBINS / 2));
    b = b < 0 ? 0 : (b > NBINS - 1 ? NBINS - 1 : b);
    bins[i] = b;
    atomicAdd(&hist[b], 1);
}

__global__ void threshold_kernel(const int* __restrict__ hist, int K,
                                 int* __restrict__ thr) {
    if (threadIdx.x != 0 || blockIdx.x != 0) return;
    int cum = 0, T = 0, ties = K;
    for (int b = NBINS - 1; b >= 0; --b) {
        int c = hist[b];
        if (cum + c >= K) { T = b; ties = K - cum; break; }
        cum += c;
    }
    thr[0] = T;
    thr[1] = ties;
}

// per-block packed (gt<<16 | eq) totals; both global totals < 65536 for N=50k
__global__ void scan_blocks_kernel(const int* __restrict__ bins,
                                   const int* __restrict__ thr,
                                   int* __restrict__ blk, int n) {
    __shared__ int s[256];
    int t = threadIdx.x;
    int i = blockIdx.x * 256 + t;
    int T = thr[0];
    int v = 0;
    if (i < n) {
        int b = bins[i];
        v = ((b > T) ? (1 << 16) : 0) | ((b == T) ? 1 : 0);
    }
    s[t] = v;
    __syncthreads();
    for (int o = 128; o > 0; o >>= 1) {
        if (t < o) s[t] += s[t + o];
        __syncthreads();
    }
    if (t == 0) blk[blockIdx.x] = s[0];
}

__global__ void scan_offsets_kernel(const int* __restrict__ blk,
                                    int* __restrict__ blkOff, int nb) {
    __shared__ int s[256];
    int t = threadIdx.x;
    int v = (t < nb) ? blk[t] : 0;
    s[t] = v;
    __syncthreads();
    for (int o = 1; o < 256; o <<= 1) {
        int x = (t >= o) ? s[t - o] : 0;
        __syncthreads();
        s[t] += x;
        __syncthreads();
    }
    if (t < nb) blkOff[t] = s[t] - v;   // exclusive
}

__global__ void select_nodes_kernel(const int* __restrict__ bins,
                                    const float* __restrict__ scores,
                                    const int* __restrict__ thr,
                                    const int* __restrict__ blkOff,
                                    int* __restrict__ nmap,
                                    int* __restrict__ perm,
                                    float* __restrict__ vals, int n) {
    __shared__ int s[256];
    int t = threadIdx.x;
    int i = blockIdx.x * 256 + t;
    int T = thr[0], ties = thr[1];
    int gt = 0, eq = 0, v = 0;
    if (i < n) {
        int b = bins[i];
        gt = (b > T);
        eq = (b == T);
        v = (gt << 16) | eq;
    }
    s[t] = v;
    __syncthreads();
    for (int o = 1; o < 256; o <<= 1) {
        int x = (t >= o) ? s[t - o] : 0;
        __syncthreads();
        s[t] += x;
        __syncthreads();
    }
    int exc = s[t] - v;
    int off = blkOff[blockIdx.x];
    int GT = (off >> 16) + (exc >> 16);
    int EQ = (off & 0xFFFF) + (exc & 0xFFFF);
    bool sel = (i < n) && (gt || (eq && EQ < ties));
    if (sel) {
        int eqs = EQ < ties ? EQ : ties;
        int pos = GT + eqs;
        nmap[i] = pos;
        perm[pos] = i;
        vals[pos] = scores[i];
    }
}

__global__ void gather_hk_kernel(const float* __restrict__ h3,
                                 const int* __restrict__ perm,
                                 const float* __restrict__ vals,
                                 float* __restrict__ hk, int K) {
    long long idx = (long long)blockIdx.x * blockDim.x + threadIdx.x;
    if (idx >= (long long)K * 32) return;
    int p = (int)(idx >> 5);
    int c = (int)(idx & 31) * 4;
    int i = perm[p];
    float g = vals[p];
    float4 v = *(const float4*)(h3 + (long long)i * 128 + c);
    float4 o;
    o.x = v.x * g; o.y = v.y * g; o.z = v.z * g; o.w = v.w * g;
    *(float4*)(hk + (long long)p * 128 + c) = o;
}

// ------------------------------ final pooling ------------------------------
__global__ void cluster_pool_kernel(const float* __restrict__ h4,
                                    float* __restrict__ pooled, int K) {
    int t = blockIdx.x * blockDim.x + threadIdx.x;   // 512 total
    if (t >= 512) return;
    int col = t & 255, c = t >> 8;
    float s = 0.f;
    int cnt = 0;
    for (int r = c; r < K; r += 2) { s += h4[(long long)r * 256 + col]; ++cnt; }
    pooled[c * 256 + col] = s / (float)cnt;
}

__global__ void final_out_kernel(const float* __restrict__ pooled,
                                 const float* __restrict__ wout,
                                 const float* __restrict__ bout,
                                 float* __restrict__ out) {
    __shared__ float s[128];
    int t = threadIdx.x;
    float a = 0.f;
    for (int j = t; j < 512; j += 128) a += pooled[j] * wout[j];
    s[t] = a;
    __syncthreads();
    for (int o = 64; o > 0; o >>= 1) {
        if (t < o) s[t] += s[t + o];
        __syncthreads();
    }
    if (t == 0) out[0] = s[0] + bout[0];
}

// ---------------------------------------------------------------------------
extern "C" void kernel_launch(void* const* d_in, const int* in_sizes, int n_in,
                              void* d_out, int out_size, void* d_ws, size_t ws_size,
                              hipStream_t stream) {
    const float* x       = (const float*)d_in[0];
    const int*   eidx    = (const int*)d_in[1];
    const float* wd_root = (const float*)d_in[2];
    const float* wd_b    = (const float*)d_in[3];
    const float* w0_rel  = (const float*)d_in[4];
    const float* w0_root = (const float*)d_in[5];
    const float* w0_b    = (const float*)d_in[6];
    const float* w1_rel  = (const float*)d_in[7];
    const float* w1_root = (const float*)d_in[8];
    const float* w1_b    = (const float*)d_in[9];
    const float* w2_rel  = (const float*)d_in[10];
    const float* w2_root = (const float*)d_in[11];
    const float* w2_b    = (const float*)d_in[12];
    const float* pool_p  = (const float*)d_in[13];
    const float* w3_rel  = (const float*)d_in[14];
    const float* w3_root = (const float*)d_in[15];
    const float* w3_b    = (const float*)d_in[16];
    const float* w_out   = (const float*)d_in[17];
    const float* b_out   = (const float*)d_in[18];
    float* out = (float*)d_out;

    const int N = in_sizes[0] / 64;
    const int E = in_sizes[1] / 2;
    const int K = (3 * N + 4) / 5;            // ceil(0.6*N)
    const int* src = eidx;
    const int* dst = eidx + E;
    const int SB = (N + 255) / 256;           // scan blocks (<=256 for N<=65536)

    // ---- carve workspace ----
    char* w = (char*)d_ws;
    auto carve = [&](size_t bytes) -> void* {
        void* p = (void*)w;
        w += (bytes + 255) & ~(size_t)255;
        return p;
    };
    float* h0     = (float*)carve((size_t)N * 4 * 4);
    float* h1     = (float*)carve((size_t)N * 32 * 4);
    float* h2     = (float*)carve((size_t)N * 64 * 4);
    float* h3     = (float*)carve((size_t)N * 128 * 4);
    float* agg    = (float*)carve((size_t)N * 128 * 4);   // reused per conv
    float* hk     = (float*)carve((size_t)K * 128 * 4);
    float* h4     = (float*)carve((size_t)K * 256 * 4);
    float* scores = (float*)carve((size_t)N * 4);
    float* vals   = (float*)carve((size_t)K * 4);
    float* pooled = (float*)carve(512 * 4);
    float* nrm    = (float*)carve(4);
    int* bins     = (int*)carve((size_t)N * 4);
    int* nmap     = (int*)carve((size_t)N * 4);
    int* perm     = (int*)carve((size_t)K * 4);
    int* hist     = (int*)carve(NBINS * 4);
    int* thr      = (int*)carve(2 * 4);
    int* blk      = (int*)carve((size_t)SB * 4);
    int* blkOff   = (int*)carve((size_t)SB * 4);
    (void)ws_size; (void)n_in; (void)out_size;

    const int T256 = 256;
    auto blocks = [](long long work, int bs) { return (int)((work + bs - 1) / bs); };

    // ---- stage 0: norm + dense input ----
    norm_kernel<<<1, 128, 0, stream>>>(pool_p, nrm);
    dense_input_kernel<<<blocks(N, T256), T256, 0, stream>>>(x, wd_root, wd_b, h0, N);

    // ---- conv0: 4 -> 32 ----
    hipMemsetAsync(agg, 0, (size_t)N * 4 * 4, stream);
    edge_agg_kernel<4><<<blocks((long long)E * 1, T256), T256, 0, stream>>>(src, dst, h0, agg, E);
    {
        int total = (N / 16) * (32 / 16);
        graphconv_gemm_kernel<4, 32><<<blocks(total, 4), 128, 0, stream>>>(
            agg, h0, w0_rel, w0_root, w0_b, h1, N);
    }

    // ---- conv1: 32 -> 64 ----
    hipMemsetAsync(agg, 0, (size_t)N * 32 * 4, stream);
    edge_agg_kernel<32><<<blocks((long long)E * 8, T256), T256, 0, stream>>>(src, dst, h1, agg, E);
    {
        int total = (N / 16) * (64 / 16);
        graphconv_gemm_kernel<32, 64><<<blocks(total, 4), 128, 0, stream>>>(
            agg, h1, w1_rel, w1_root, w1_b, h2, N);
    }

    // ---- conv2: 64 -> 128 ----
    hipMemsetAsync(agg, 0, (size_t)N * 64 * 4, stream);
    edge_agg_kernel<64><<<blocks((long long)E * 16, T256), T256, 0, stream>>>(src, dst, h2, agg, E);
    {
        int total = (N / 16) * (128 / 16);
        graphconv_gemm_kernel<64, 128><<<blocks(total, 4), 128, 0, stream>>>(
            agg, h2, w2_rel, w2_root, w2_b, h3, N);
    }

    // ---- TopK pooling ----
    score_kernel<<<blocks(N, T256), T256, 0, stream>>>(h3, pool_p, nrm, scores, N);
    hipMemsetAsync(hist, 0, NBINS * 4, stream);
    hist_kernel<<<blocks(N, T256), T256, 0, stream>>>(scores, hist, bins, N);
    threshold_kernel<<<1, 64, 0, stream>>>(hist, K, thr);
    scan_blocks_kernel<<<SB, 256, 0, stream>>>(bins, thr, blk, N);
    scan_offsets_kernel<<<1, 256, 0, stream>>>(blk, blkOff, SB);
    hipMemsetAsync(nmap, 0xFF, (size_t)N * 4, stream);   // -1
    select_nodes_kernel<<<SB, 256, 0, stream>>>(bins, scores, thr, blkOff, nmap, perm, vals, N);
    gather_hk_kernel<<<blocks((long long)K * 32, T256), T256, 0, stream>>>(h3, perm, vals, hk, K);

    // ---- conv3 on filtered graph: 128 -> 256 ----
    hipMemsetAsync(agg, 0, (size_t)K * 128 * 4, stream);
    edge_agg_filtered_kernel<<<blocks((long long)E * 32, T256), T256, 0, stream>>>(
        src, dst, nmap, hk, agg, E);
    {
        int total = (K / 16) * (256 / 16);
        graphconv_gemm_kernel<128, 256><<<blocks(total, 4), 128, 0, stream>>>(
            agg, hk, w3_rel, w3_root, w3_b, h4, K);
    }

    // ---- cluster mean pool + readout ----
    cluster_pool_kernel<<<2, 256, 0, stream>>>(h4, pooled, K);
    final_out_kernel<<<1, 128, 0, stream>>>(pooled, w_out, b_out, out);
}